// RoEmbedding_57423712747840
// MI455X (gfx1250) — compile-verified
//
#include <hip/hip_runtime.h>

// Problem constants (reference: B,S,D,INNER = 4,2048,1024,64)
#define BB 4
#define SS 2048
#define DD 1024
#define NOUT 256          // INNER*4 projection outputs
#define INNER_D 64

typedef __attribute__((ext_vector_type(2))) float v2f;
typedef __attribute__((ext_vector_type(8))) float v8f;

// D = A(16x4 f32) * B(4x16 f32) + C(16x16 f32)
__device__ __forceinline__ v8f wmma_f32x4(v2f a, v2f b, v8f c) {
    return __builtin_amdgcn_wmma_f32_16x16x4_f32(
        /*neg_a=*/false, a, /*neg_b=*/false, b,
        /*c_mod=*/(short)0, c, /*reuse_a=*/false, /*reuse_b=*/false);
}

// ---------------------------------------------------------------------------
// Kernel 1: out = X @ W^T + b, split into q/k, apply RoPE, store to ws.
// Block = 256 threads (8 waves). Block computes 64 rows x 256 cols.
// Wave w: M-tile mi = w&3 (16 rows), N-half = w>>2 (8 N-tiles of 16).
// ---------------------------------------------------------------------------
__global__ __launch_bounds__(256) void proj_rope_kernel(
    const float* __restrict__ X,     // (B*S, 1024)
    const float* __restrict__ W,     // (256, 1024)
    const float* __restrict__ bias,  // (256)
    float* __restrict__ Qws,         // (B,2,S,64)
    float* __restrict__ Kws)         // (B,2,S,64)
{
    const int lane  = threadIdx.x & 31;
    const int wave  = threadIdx.x >> 5;
    const int mi    = wave & 3;
    const int nhalf = wave >> 2;
    const int lm    = lane & 15;   // M (or N) within tile
    const int g     = lane >> 4;   // K sub-group (0/1)

    const int m0g = blockIdx.x * 64 + mi * 16;           // global row base
    const float* arow = X + (size_t)(m0g + lm) * DD + 2 * g;

    const float* wrow[8];
#pragma unroll
    for (int ni = 0; ni < 8; ++ni)
        wrow[ni] = W + (size_t)(nhalf * 128 + ni * 16 + lm) * DD + 2 * g;

    v8f acc[8] = {};
#pragma unroll 2
    for (int k0 = 0; k0 < DD; k0 += 4) {
        v2f a = *(const v2f*)(arow + k0);
        v2f bf[8];
#pragma unroll
        for (int ni = 0; ni < 8; ++ni) bf[ni] = *(const v2f*)(wrow[ni] + k0);
#pragma unroll
        for (int ni = 0; ni < 8; ++ni) acc[ni] = wmma_f32x4(a, bf[ni], acc[ni]);
    }

    // Epilogue: +bias, RoPE, scatter into Q/K  (no divergence before shuffles)
#pragma unroll
    for (int ni = 0; ni < 8; ++ni) {
        const int n = nhalf * 128 + ni * 16 + lm;       // 0..255
        const float bv = bias[n];
        const int h   = n >> 7;
        const int rem = n & 127;
        const int isK = (rem >= 64) ? 1 : 0;
        const int d   = rem & 63;
        const int j   = d >> 1;
        const float inv = __powf(15.0f, -(float)j * (2.0f / (float)INNER_D));
#pragma unroll
        for (int r = 0; r < 8; ++r) {
            const int row = m0g + r + 8 * g;            // global row = b*S + s
            const int s   = row & (SS - 1);
            const int bq  = row >> 11;
            float val = acc[ni][r] + bv;
            float partner = __shfl_xor(val, 1, 32);     // element d^1 (same m,h,q/k)
            float sn, cs;
            __sincosf((float)s * inv, &sn, &cs);
            float rv = (d & 1) ? (val * cs + partner * sn)
                               : (val * cs - partner * sn);
            float* dst = (isK ? Kws : Qws) +
                         (((size_t)bq * 2 + h) * SS + s) * INNER_D + d;
            *dst = rv;
        }
    }
}

// ---------------------------------------------------------------------------
// Kernel 2: logits = (qw @ kw^T) * sim for both heads, write pred_logits,
// masked NLL partial per wave.  Block = 128 threads (4 waves).
// Wave computes 2 heads x 32 M x 64 N (2 M-tiles x 4 N-tiles).
// grid = (S/64, S/128, B) = (32, 16, 4)
// Stream traffic (sim/labels/masks reads, pred writes) uses non-temporal
// cache policy so the L2-resident Q/K working set is not evicted.
// ---------------------------------------------------------------------------
__global__ __launch_bounds__(128) void logits_loss_kernel(
    const float* __restrict__ Qws, const float* __restrict__ Kws,
    const float* __restrict__ sim, const int* __restrict__ labels,
    const int* __restrict__ masks, float* __restrict__ pred,
    float* __restrict__ psum, unsigned int* __restrict__ pcnt)
{
    const int lane = threadIdx.x & 31;
    const int wave = threadIdx.x >> 5;
    const int b  = blockIdx.z;
    const int n0 = blockIdx.x * 64;
    const int m0 = (blockIdx.y * 4 + wave) * 32;
    const int lm = lane & 15, g = lane >> 4;

    const float* qb = Qws + (size_t)b * 2 * SS * INNER_D;
    const float* kb = Kws + (size_t)b * 2 * SS * INNER_D;

    const float* arow[2][2];   // [head][mi]
    const float* brow[2][4];   // [head][ni]
#pragma unroll
    for (int h = 0; h < 2; ++h) {
#pragma unroll
        for (int mi2 = 0; mi2 < 2; ++mi2)
            arow[h][mi2] = qb + ((size_t)h * SS + m0 + mi2 * 16 + lm) * INNER_D + 2 * g;
#pragma unroll
        for (int ni = 0; ni < 4; ++ni)
            brow[h][ni] = kb + ((size_t)h * SS + n0 + ni * 16 + lm) * INNER_D + 2 * g;
    }

    v8f acc[2][2][4] = {};
#pragma unroll 2
    for (int k0 = 0; k0 < INNER_D; k0 += 4) {
        v2f a[2][2], bf[2][4];
#pragma unroll
        for (int h = 0; h < 2; ++h) {
#pragma unroll
            for (int mi2 = 0; mi2 < 2; ++mi2) a[h][mi2] = *(const v2f*)(arow[h][mi2] + k0);
#pragma unroll
            for (int ni = 0; ni < 4; ++ni)  bf[h][ni] = *(const v2f*)(brow[h][ni] + k0);
        }
#pragma unroll
        for (int h = 0; h < 2; ++h)
#pragma unroll
            for (int mi2 = 0; mi2 < 2; ++mi2)
#pragma unroll
                for (int ni = 0; ni < 4; ++ni)
                    acc[h][mi2][ni] = wmma_f32x4(a[h][mi2], bf[h][ni], acc[h][mi2][ni]);
    }

    // Epilogue: scale by similarity, emit logits, masked NLL
    float lsum = 0.0f;
    unsigned int cnt = 0;
#pragma unroll
    for (int mi2 = 0; mi2 < 2; ++mi2) {
#pragma unroll
        for (int ni = 0; ni < 4; ++ni) {
#pragma unroll
            for (int r = 0; r < 8; ++r) {
                const int m = m0 + mi2 * 16 + r + 8 * g;
                const int n = n0 + ni * 16 + lm;
                const size_t idx = ((size_t)(b * SS + m)) * SS + n;
                const float sv = __builtin_nontemporal_load(sim + idx);
                const float l0 = acc[0][mi2][ni][r] * sv;
                const float l1 = acc[1][mi2][ni][r] * sv;
                // d_out+1 is only 4B-aligned -> 2x b32 stores, NT policy
                __builtin_nontemporal_store(l0, pred + idx * 2);
                __builtin_nontemporal_store(l1, pred + idx * 2 + 1);
                const float mx  = fmaxf(l0, l1);
                const float lse = mx + __logf(__expf(l0 - mx) + __expf(l1 - mx));
                const int   lab = __builtin_nontemporal_load(labels + idx);
                const int   msk = __builtin_nontemporal_load(masks + idx);
                const float lsel = lab ? l1 : l0;
                if (msk == 1) { lsum += lse - lsel; cnt++; }
            }
        }
    }

    // wave32 reduction -> one partial per wave (deterministic, no atomics)
#pragma unroll
    for (int off = 16; off; off >>= 1) {
        lsum += __shfl_xor(lsum, off, 32);
        cnt  += (unsigned int)__shfl_xor((int)cnt, off, 32);
    }
    if (lane == 0) {
        const int wid = (((int)blockIdx.z * 16 + (int)blockIdx.y) * 32 + (int)blockIdx.x) * 4 + wave;
        psum[wid] = lsum;
        pcnt[wid] = cnt;
    }
}

// ---------------------------------------------------------------------------
// Kernel 3: deterministic tree reduction of 8192 partials -> loss scalar
// ---------------------------------------------------------------------------
__global__ __launch_bounds__(256) void reduce_kernel(
    const float* __restrict__ psum, const unsigned int* __restrict__ pcnt,
    float* __restrict__ out)
{
    __shared__ double ssum[256];
    __shared__ unsigned int scnt[256];
    double s = 0.0; unsigned int c = 0;
    for (int i = threadIdx.x; i < 8192; i += 256) { s += (double)psum[i]; c += pcnt[i]; }
    ssum[threadIdx.x] = s; scnt[threadIdx.x] = c;
    __syncthreads();
    for (int off = 128; off; off >>= 1) {
        if ((int)threadIdx.x < off) {
            ssum[threadIdx.x] += ssum[threadIdx.x + off];
            scnt[threadIdx.x] += scnt[threadIdx.x + off];
        }
        __syncthreads();
    }
    if (threadIdx.x == 0) out[0] = (float)(ssum[0] / (double)scnt[0]);
}

// ---------------------------------------------------------------------------
extern "C" void kernel_launch(void* const* d_in, const int* in_sizes, int n_in,
                              void* d_out, int out_size, void* d_ws, size_t ws_size,
                              hipStream_t stream) {
    const float* X      = (const float*)d_in[0];  // sequence_outputs (B,S,D)
    const float* W      = (const float*)d_in[1];  // (256, 1024)
    const float* bias   = (const float*)d_in[2];  // (256)
    const float* sim    = (const float*)d_in[3];  // (B,S,S,1)
    const int*   labels = (const int*)d_in[4];    // (B,S,S)
    const int*   masks  = (const int*)d_in[5];    // (B,S,S)

    float* out  = (float*)d_out;
    float* pred = out + 1;                        // (loss, pred_logits) flat

    // Workspace layout: [psum 8192 f32][pcnt 8192 u32][Q 4MB][K 4MB]
    char* ws = (char*)d_ws;
    float*        psum = (float*)ws;
    unsigned int* pcnt = (unsigned int*)(ws + 8192 * sizeof(float));
    float* Qws = (float*)(ws + 8192 * (sizeof(float) + sizeof(unsigned int)));
    float* Kws = Qws + (size_t)BB * 2 * SS * INNER_D;

    proj_rope_kernel<<<dim3((BB * SS) / 64), dim3(256), 0, stream>>>(
        X, W, bias, Qws, Kws);
    logits_loss_kernel<<<dim3(SS / 64, SS / 128, BB), dim3(128), 0, stream>>>(
        Qws, Kws, sim, labels, masks, pred, psum, pcnt);
    reduce_kernel<<<dim3(1), dim3(256), 0, stream>>>(psum, pcnt, out);
}